// LRU_28819230556516
// MI455X (gfx1250) — compile-verified
//
#include <hip/hip_runtime.h>

// ---------------------------------------------------------------------------
// LRU forward on MI455X (gfx1250, wave32).
//   Phase 1: Bu = x @ B_norm^T      (2 real GEMMs, V_WMMA_F32_16X16X4_F32)
//   Phase 2: chunked diagonal complex scan over T (in-place on Bu planes)
//   Phase 3: y  = Re(states @ C^T)  (2 real GEMMs fused into one accumulator)
//
// fp32 WMMA keeps reference precision; the problem is HBM-bandwidth-dominated
// (~0.5-1 GB moved vs 34 GFLOP at 23.3 TB/s), so no payoff for f16/bf16.
//
// Weight planes are pre-packed so each WMMA B-fragment is one aligned
// global_load_b64:  wp[(k>>1)*2N + col*2 + (k&1)] = W^T[k][col].
// Each wave computes a 16(M)x32(N) tile: the A fragment feeds 4 WMMAs.
// ---------------------------------------------------------------------------

typedef __attribute__((ext_vector_type(2))) float v2f;
typedef __attribute__((ext_vector_type(8))) float v8f;

static constexpr int kB      = 8;
static constexpr int kT      = 8192;
static constexpr int kN      = 256;
static constexpr int kBT     = kB * kT;       // 65536 GEMM rows
static constexpr int kChunk  = 512;
static constexpr int kNChunk = kT / kChunk;   // 16

__device__ __forceinline__ v8f wmma_f32(v2f a, v2f b, v8f c) {
  // D = A(16x4 f32) x B(4x16 f32) + C(16x16 f32)
  return __builtin_amdgcn_wmma_f32_16x16x4_f32(
      /*neg_a=*/false, a, /*neg_b=*/false, b,
      /*c_mod=*/(short)0, c, /*reuse_a=*/false, /*reuse_b=*/false);
}

// Packed-pair index for K-major weights: (k, col) -> flat float offset.
__device__ __forceinline__ size_t wpack_idx(int k, int col) {
  return (size_t)(k >> 1) * (2 * kN) + col * 2 + (k & 1);
}

// ---------------------------------------------------------------------------
// Prep: Lambda diag + packed K-major weight planes. grid=(kN), block=(kN).
//   wb*_p[(n>>1)*2N + m*2 + (n&1)] = B_*[m][n]*exp(gamma_log[m])
//   wcre_p: C_re[m][n];  wcim_np: -C_im[m][n]
// ---------------------------------------------------------------------------
__global__ void __launch_bounds__(kN)
lru_prep(const float* __restrict__ nu_log, const float* __restrict__ theta_log,
         const float* __restrict__ gamma_log,
         const float* __restrict__ B_re, const float* __restrict__ B_im,
         const float* __restrict__ C_re, const float* __restrict__ C_im,
         float* __restrict__ wbre_p, float* __restrict__ wbim_p,
         float* __restrict__ wcre_p, float* __restrict__ wcim_np,
         float* __restrict__ l_re, float* __restrict__ l_im) {
  const int m = blockIdx.x;    // weight row (output index of einsum)
  const int n = threadIdx.x;   // weight col (contraction index k)
  const float g = expf(gamma_log[m]);
  const size_t p = wpack_idx(n, m);
  wbre_p[p]  = B_re[m * kN + n] * g;
  wbim_p[p]  = B_im[m * kN + n] * g;
  wcre_p[p]  = C_re[m * kN + n];
  wcim_np[p] = -C_im[m * kN + n];
  if (m == 0) {
    const float lmod = expf(-expf(nu_log[n]));
    const float th   = expf(theta_log[n]);
    l_re[n] = lmod * cosf(th);
    l_im[n] = lmod * sinf(th);
  }
}

// ---------------------------------------------------------------------------
// GEMM 1: one wave -> 16x32 tile of (Bu_re, Bu_im); A fragment feeds 4 WMMAs.
// grid = (kBT/16)*(kN/32) waves / 8 waves per block = 4096 blocks.
// A layout (ISA 16x4 f32): lane&15 = M row, lane>>4 selects K pair.
// ---------------------------------------------------------------------------
__global__ void __launch_bounds__(256)
lru_gemm_in(const float* __restrict__ x,
            const float* __restrict__ wre_p, const float* __restrict__ wim_p,
            float* __restrict__ out_re, float* __restrict__ out_im) {
  const int lane = threadIdx.x & 31;
  const int wid  = blockIdx.x * 8 + (threadIdx.x >> 5);
  const int nt2  = wid & 7;     // 8 pairs of n-tiles (32 cols each)
  const int mt   = wid >> 3;    // 4096 tiles over rows (b*t)
  const int lm   = lane & 15;
  const int lhi  = lane >> 4;
  const int col0 = nt2 * 32 + lm;
  const int col1 = col0 + 16;

  const float* arow = x + (size_t)(mt * 16 + lm) * kN;
  v8f acc_re0 = {}, acc_re1 = {};
  v8f acc_im0 = {}, acc_im1 = {};
#pragma unroll 4
  for (int k0 = 0; k0 < kN; k0 += 4) {
    const int kb = k0 + lhi * 2;                       // even
    const v2f a = *(const v2f*)(arow + kb);            // 8B aligned
    const v2f bre0 = *(const v2f*)(wre_p + wpack_idx(kb, col0));
    const v2f bre1 = *(const v2f*)(wre_p + wpack_idx(kb, col1));
    const v2f bim0 = *(const v2f*)(wim_p + wpack_idx(kb, col0));
    const v2f bim1 = *(const v2f*)(wim_p + wpack_idx(kb, col1));
    acc_re0 = wmma_f32(a, bre0, acc_re0);
    acc_re1 = wmma_f32(a, bre1, acc_re1);
    acc_im0 = wmma_f32(a, bim0, acc_im0);
    acc_im1 = wmma_f32(a, bim1, acc_im1);
  }
#pragma unroll
  for (int r = 0; r < 8; ++r) {
    const int row = mt * 16 + r + lhi * 8;             // C/D: halves split M
    out_re[(size_t)row * kN + col0] = acc_re0[r];
    out_re[(size_t)row * kN + col1] = acc_re1[r];
    out_im[(size_t)row * kN + col0] = acc_im0[r];
    out_im[(size_t)row * kN + col1] = acc_im1[r];
  }
}

// ---------------------------------------------------------------------------
// Scan pass A: in-place local inclusive scan per chunk; emit chunk carries.
// grid = kB*kNChunk blocks, 256 threads (one per n) -> coalesced per step.
// ---------------------------------------------------------------------------
__global__ void __launch_bounds__(kN)
lru_scan_local(float* __restrict__ bu_re, float* __restrict__ bu_im,
               const float* __restrict__ l_re, const float* __restrict__ l_im,
               float* __restrict__ carry_re, float* __restrict__ carry_im) {
  const int b = blockIdx.x >> 4;     // kNChunk == 16
  const int c = blockIdx.x & 15;
  const int n = threadIdx.x;
  const float lr = l_re[n], li = l_im[n];
  float hr = 0.f, hi = 0.f;
  const size_t base = ((size_t)b * kT + (size_t)c * kChunk) * kN + n;
  for (int t = 0; t < kChunk; ++t) {
    const size_t idx = base + (size_t)t * kN;
    const float br = bu_re[idx];
    const float bi = bu_im[idx];
    const float nr = lr * hr - li * hi + br;   // h = L*h + Bu_t
    const float ni = lr * hi + li * hr + bi;
    bu_re[idx] = nr;
    bu_im[idx] = ni;
    hr = nr; hi = ni;
  }
  carry_re[(size_t)blockIdx.x * kN + n] = hr;
  carry_im[(size_t)blockIdx.x * kN + n] = hi;
}

// ---------------------------------------------------------------------------
// Scan pass B: exclusive scan of carries across chunks. 1 block, 256 threads.
// Chunk transition factor is L^kChunk (9 complex squarings).
// ---------------------------------------------------------------------------
__global__ void __launch_bounds__(kN)
lru_scan_carry(const float* __restrict__ l_re, const float* __restrict__ l_im,
               const float* __restrict__ carry_re, const float* __restrict__ carry_im,
               float* __restrict__ pre_re, float* __restrict__ pre_im) {
  const int n = threadIdx.x;
  float ar = l_re[n], ai = l_im[n];
#pragma unroll
  for (int i = 0; i < 9; ++i) {      // L^(2^9) = L^512 = L^kChunk
    const float nr = ar * ar - ai * ai;
    const float ni = 2.f * ar * ai;
    ar = nr; ai = ni;
  }
  for (int b = 0; b < kB; ++b) {
    float pr = 0.f, pi = 0.f;
    for (int c = 0; c < kNChunk; ++c) {
      const size_t idx = ((size_t)b * kNChunk + c) * kN + n;
      pre_re[idx] = pr;
      pre_im[idx] = pi;
      const float cr = carry_re[idx];
      const float ci = carry_im[idx];
      const float nr = ar * pr - ai * pi + cr;
      const float ni = ar * pi + ai * pr + ci;
      pr = nr; pi = ni;
    }
  }
}

// ---------------------------------------------------------------------------
// Scan pass C: states_t += L^(t_local+1) * prefix. Chunk 0 has zero prefix
// and is skipped. grid = kB*(kNChunk-1).
// ---------------------------------------------------------------------------
__global__ void __launch_bounds__(kN)
lru_scan_apply(float* __restrict__ bu_re, float* __restrict__ bu_im,
               const float* __restrict__ l_re, const float* __restrict__ l_im,
               const float* __restrict__ pre_re, const float* __restrict__ pre_im) {
  const int b = blockIdx.x / (kNChunk - 1);
  const int c = 1 + blockIdx.x % (kNChunk - 1);
  const int n = threadIdx.x;
  const size_t cidx = ((size_t)b * kNChunk + c) * kN + n;
  const float pr = pre_re[cidx], pi = pre_im[cidx];
  const float lr = l_re[n], li = l_im[n];
  float qr = lr * pr - li * pi;      // q = L^1 * prefix
  float qi = lr * pi + li * pr;
  const size_t base = ((size_t)b * kT + (size_t)c * kChunk) * kN + n;
  for (int t = 0; t < kChunk; ++t) {
    const size_t idx = base + (size_t)t * kN;
    bu_re[idx] += qr;
    bu_im[idx] += qi;
    const float nr = lr * qr - li * qi;
    const float ni = lr * qi + li * qr;
    qr = nr; qi = ni;
  }
}

// ---------------------------------------------------------------------------
// GEMM 2: y = s_re @ Cre^T + s_im @ (-Cim)^T. One wave -> 16x32 tile,
// two accumulators (one per n-tile), 4 WMMAs per k-step.
// ---------------------------------------------------------------------------
__global__ void __launch_bounds__(256)
lru_gemm_out(const float* __restrict__ s_re, const float* __restrict__ s_im,
             const float* __restrict__ wre_p, const float* __restrict__ wimn_p,
             float* __restrict__ y) {
  const int lane = threadIdx.x & 31;
  const int wid  = blockIdx.x * 8 + (threadIdx.x >> 5);
  const int nt2  = wid & 7;
  const int mt   = wid >> 3;
  const int lm   = lane & 15;
  const int lhi  = lane >> 4;
  const int col0 = nt2 * 32 + lm;
  const int col1 = col0 + 16;

  const float* arow_re = s_re + (size_t)(mt * 16 + lm) * kN;
  const float* arow_im = s_im + (size_t)(mt * 16 + lm) * kN;
  v8f acc0 = {}, acc1 = {};
#pragma unroll 4
  for (int k0 = 0; k0 < kN; k0 += 4) {
    const int kb = k0 + lhi * 2;
    const v2f a_re = *(const v2f*)(arow_re + kb);
    const v2f a_im = *(const v2f*)(arow_im + kb);
    const v2f bre0 = *(const v2f*)(wre_p  + wpack_idx(kb, col0));
    const v2f bre1 = *(const v2f*)(wre_p  + wpack_idx(kb, col1));
    const v2f bim0 = *(const v2f*)(wimn_p + wpack_idx(kb, col0));
    const v2f bim1 = *(const v2f*)(wimn_p + wpack_idx(kb, col1));
    acc0 = wmma_f32(a_re, bre0, acc0);
    acc1 = wmma_f32(a_re, bre1, acc1);
    acc0 = wmma_f32(a_im, bim0, acc0);   // -C_im folded in at prep
    acc1 = wmma_f32(a_im, bim1, acc1);
  }
#pragma unroll
  for (int r = 0; r < 8; ++r) {
    const int row = mt * 16 + r + lhi * 8;
    y[(size_t)row * kN + col0] = acc0[r];
    y[(size_t)row * kN + col1] = acc1[r];
  }
}

// ---------------------------------------------------------------------------
extern "C" void kernel_launch(void* const* d_in, const int* in_sizes, int n_in,
                              void* d_out, int out_size, void* d_ws, size_t ws_size,
                              hipStream_t stream) {
  (void)in_sizes; (void)n_in; (void)out_size; (void)ws_size;

  const float* x         = (const float*)d_in[0];
  const float* nu_log    = (const float*)d_in[1];
  const float* theta_log = (const float*)d_in[2];
  const float* gamma_log = (const float*)d_in[3];
  const float* B_re      = (const float*)d_in[4];
  const float* B_im      = (const float*)d_in[5];
  const float* C_re      = (const float*)d_in[6];
  const float* C_im      = (const float*)d_in[7];
  float* out = (float*)d_out;

  // Workspace carve-up (floats). Total ~33.8M floats (~135 MB).
  float* ws = (float*)d_ws;
  const size_t plane = (size_t)kBT * kN;          // 16,777,216
  float* bu_re    = ws;
  float* bu_im    = bu_re + plane;
  float* wbre_p   = bu_im + plane;
  float* wbim_p   = wbre_p + (size_t)kN * kN;
  float* wcre_p   = wbim_p + (size_t)kN * kN;
  float* wcim_np  = wcre_p + (size_t)kN * kN;
  float* l_re     = wcim_np + (size_t)kN * kN;
  float* l_im     = l_re + kN;
  float* carry_re = l_im + kN;
  float* carry_im = carry_re + (size_t)kB * kNChunk * kN;
  float* pre_re   = carry_im + (size_t)kB * kNChunk * kN;
  float* pre_im   = pre_re + (size_t)kB * kNChunk * kN;

  // 1) Lambda + packed weight planes
  lru_prep<<<kN, kN, 0, stream>>>(nu_log, theta_log, gamma_log,
                                  B_re, B_im, C_re, C_im,
                                  wbre_p, wbim_p, wcre_p, wcim_np, l_re, l_im);

  // 2) Bu = x @ B_norm^T  (65536x256x256, re+im)
  const int total_waves = (kBT / 16) * (kN / 32);   // 32768 (16x32 tiles)
  lru_gemm_in<<<total_waves / 8, 256, 0, stream>>>(x, wbre_p, wbim_p, bu_re, bu_im);

  // 3) chunked diagonal scan over T (in place)
  lru_scan_local<<<kB * kNChunk, kN, 0, stream>>>(bu_re, bu_im, l_re, l_im,
                                                  carry_re, carry_im);
  lru_scan_carry<<<1, kN, 0, stream>>>(l_re, l_im, carry_re, carry_im,
                                       pre_re, pre_im);
  lru_scan_apply<<<kB * (kNChunk - 1), kN, 0, stream>>>(bu_re, bu_im, l_re, l_im,
                                                        pre_re, pre_im);

  // 4) y = Re(states @ C^T)
  lru_gemm_out<<<total_waves / 8, 256, 0, stream>>>(bu_re, bu_im,
                                                    wcre_p, wcim_np, out);
}